// NoNorm_49847390437779
// MI455X (gfx1250) — compile-verified
//
#include <hip/hip_runtime.h>
#include <hip/hip_bf16.h>

typedef __attribute__((ext_vector_type(4))) float f4;
typedef __attribute__((ext_vector_type(2))) float v2f;
typedef __attribute__((ext_vector_type(8))) float v8f;

#define BATCH 64
#define CHAN  256
#define HW    3136          // 56*56
#define PLANE4 784          // HW/4 float4 per (b,c) plane
static constexpr long XN = (long)BATCH * CHAN * HW;   // 51,380,224

// ---------------------------------------------------------------------------
// Kernel 1 (dominant, memory-bound): stream x -> copy to out, and produce
// per-(b,c) sum / sum-of-squares. One block per 3136-float plane.
// Non-temporal hints: 411 MB streamed once, > 192 MB L2, never re-read.
// ---------------------------------------------------------------------------
__global__ void __launch_bounds__(256)
reduce_copy_kernel(const float* __restrict__ x, float* __restrict__ out,
                   float* __restrict__ s, float* __restrict__ ss) {
    __shared__ float red_s[256];
    __shared__ float red_q[256];
    const int plane = blockIdx.x;                 // == b*CHAN + c
    const long base4 = (long)plane * PLANE4;
    const f4* __restrict__ x4 = (const f4*)x;
    f4* __restrict__ o4 = (f4*)out;
    float acc_s = 0.f, acc_q = 0.f;
    for (int i = threadIdx.x; i < PLANE4; i += 256) {
        f4 v = __builtin_nontemporal_load(&x4[base4 + i]);
        __builtin_nontemporal_store(v, &o4[base4 + i]);
        acc_s += (v.x + v.y) + (v.z + v.w);
        acc_q += (v.x * v.x + v.y * v.y) + (v.z * v.z + v.w * v.w);
    }
    red_s[threadIdx.x] = acc_s;
    red_q[threadIdx.x] = acc_q;
    __syncthreads();
    for (int off = 128; off > 0; off >>= 1) {
        if ((int)threadIdx.x < off) {
            red_s[threadIdx.x] += red_s[threadIdx.x + off];
            red_q[threadIdx.x] += red_q[threadIdx.x + off];
        }
        __syncthreads();
    }
    if (threadIdx.x == 0) {
        s[plane]  = red_s[0];
        ss[plane] = red_q[0];
    }
}

// ---------------------------------------------------------------------------
// Kernel 2: BatchNorm stats via WMMA. mean_bn[c] = sum_b s[b][c] / (B*HW).
// ones(16x4) x S(4x16) chained over 16 K-chunks => D columns = batch sums.
// One wave (32 lanes) per 16 channels; 16 blocks cover all 256 channels.
// ---------------------------------------------------------------------------
__global__ void __launch_bounds__(32)
bn_stats_wmma(const float* __restrict__ s, const float* __restrict__ ss,
              float* __restrict__ mean_bn, float* __restrict__ var_bn) {
    const int c0   = blockIdx.x * 16;
    const int lane = threadIdx.x;        // 0..31, full wave, EXEC all 1s
    const int n    = lane & 15;
    const int hi   = lane >> 4;          // lane half selects B-row parity
    v2f a; a.x = 1.0f; a.y = 1.0f;       // A = ones(16x4): D[m][n] = sum_k B[k][n]
    v8f cs  = {};
    v8f cq  = {};
    for (int kb = 0; kb < 16; ++kb) {
        const int b0 = kb * 4;
        // B(4x16) layout: v0 = {K0 in lanes 0-15, K1 in lanes 16-31},
        //                 v1 = {K2 in lanes 0-15, K3 in lanes 16-31}
        v2f bs, bq;
        bs.x = s [(b0 + hi)     * CHAN + c0 + n];
        bs.y = s [(b0 + 2 + hi) * CHAN + c0 + n];
        bq.x = ss[(b0 + hi)     * CHAN + c0 + n];
        bq.y = ss[(b0 + 2 + hi) * CHAN + c0 + n];
        cs = __builtin_amdgcn_wmma_f32_16x16x4_f32(false, a, false, bs,
                                                   (short)0, cs, false, false);
        cq = __builtin_amdgcn_wmma_f32_16x16x4_f32(false, a, false, bq,
                                                   (short)0, cq, false, false);
    }
    // D VGPR0: lanes 0-15 hold D[0][n] = column-n sum (all rows identical)
    const float inv_n = 1.0f / (float)((long)BATCH * HW);
    float m = cs[0] * inv_n;
    float v = cq[0] * inv_n - m * m;
    if (lane < 16) {
        mean_bn[c0 + n] = m;
        var_bn[c0 + n]  = v;
    }
}

// ---------------------------------------------------------------------------
// Kernel 3: hierarchical group stats (g = 2,4,8,16), one block per batch.
// ---------------------------------------------------------------------------
__global__ void __launch_bounds__(128)
group_stats_kernel(const float* __restrict__ s, const float* __restrict__ ss,
                   float* __restrict__ out) {
    __shared__ float ls[256];
    __shared__ float lq[256];
    const int b = blockIdx.x;
    const int t = threadIdx.x;           // 0..127
    ls[t]       = s [b * CHAN + t];
    ls[t + 128] = s [b * CHAN + t + 128];
    lq[t]       = ss[b * CHAN + t];
    lq[t + 128] = ss[b * CHAN + t + 128];
    __syncthreads();

    float* m2  = out + XN + 512;
    float* v2  = m2  + 8192;
    float* m4  = v2  + 8192;
    float* v4  = m4  + 4096;
    float* m8  = v4  + 4096;
    float* v8  = m8  + 2048;
    float* m16 = v8  + 2048;
    float* v16 = m16 + 1024;

    // g = 2 (128 outputs)
    float s2 = ls[2 * t] + ls[2 * t + 1];
    float q2 = lq[2 * t] + lq[2 * t + 1];
    { float m = s2 / 6272.0f;  m2[b * 128 + t] = m;  v2[b * 128 + t] = q2 / 6272.0f  - m * m; }
    __syncthreads();
    ls[t] = s2; lq[t] = q2;
    __syncthreads();

    // g = 4 (64 outputs)
    float s4 = 0.f, q4 = 0.f;
    if (t < 64) {
        s4 = ls[2 * t] + ls[2 * t + 1];
        q4 = lq[2 * t] + lq[2 * t + 1];
        float m = s4 / 12544.0f; m4[b * 64 + t] = m; v4[b * 64 + t] = q4 / 12544.0f - m * m;
    }
    __syncthreads();
    if (t < 64) { ls[t] = s4; lq[t] = q4; }
    __syncthreads();

    // g = 8 (32 outputs)
    float s8 = 0.f, q8 = 0.f;
    if (t < 32) {
        s8 = ls[2 * t] + ls[2 * t + 1];
        q8 = lq[2 * t] + lq[2 * t + 1];
        float m = s8 / 25088.0f; m8[b * 32 + t] = m; v8[b * 32 + t] = q8 / 25088.0f - m * m;
    }
    __syncthreads();
    if (t < 32) { ls[t] = s8; lq[t] = q8; }
    __syncthreads();

    // g = 16 (16 outputs)
    if (t < 16) {
        float sg = ls[2 * t] + ls[2 * t + 1];
        float qg = lq[2 * t] + lq[2 * t + 1];
        float m = sg / 50176.0f;
        m16[b * 16 + t] = m;
        v16[b * 16 + t] = qg / 50176.0f - m * m;
    }
}

extern "C" void kernel_launch(void* const* d_in, const int* in_sizes, int n_in,
                              void* d_out, int out_size, void* d_ws, size_t ws_size,
                              hipStream_t stream) {
    const float* x = (const float*)d_in[0];
    float* out = (float*)d_out;
    float* s   = (float*)d_ws;            // 16384 floats
    float* ss  = s + BATCH * CHAN;        // 16384 floats (128 KB total)

    // Pass 1: bandwidth-bound copy + per-(b,c) sum/sumsq
    reduce_copy_kernel<<<BATCH * CHAN, 256, 0, stream>>>(x, out, s, ss);
    // Pass 2: BN stats over batch via v_wmma_f32_16x16x4_f32
    bn_stats_wmma<<<CHAN / 16, 32, 0, stream>>>(s, ss, out + XN, out + XN + 256);
    // Pass 3: grouped stats hierarchy
    group_stats_kernel<<<BATCH, 128, 0, stream>>>(s, ss, out);
}